// Residual_28226525069323
// MI455X (gfx1250) — compile-verified
//
#include <hip/hip_runtime.h>
#include <hip/hip_bf16.h>

#define CH 128
#define BN_EPS 1e-5f

typedef __attribute__((ext_vector_type(2))) float v2f;
typedef __attribute__((ext_vector_type(8))) float v8f;

// Native CDNA5 float atomic add (no-return form -> STOREcnt, flushed at s_endpgm).
__device__ __forceinline__ void atomic_add_f32_native(float* p, float v) {
  asm volatile("global_atomic_add_f32 %0, %1, off" :: "v"(p), "v"(v) : "memory");
}

// ---------------------------------------------------------------- elementwise
__global__ void fill_kernel(float* __restrict__ p, float val, int n) {
  int i = blockIdx.x * blockDim.x + threadIdx.x;
  if (i < n) p[i] = val;
}

__global__ void degree_kernel(const int* __restrict__ dst, float* __restrict__ deg, int E) {
  int e = blockIdx.x * blockDim.x + threadIdx.x;
  if (e < E) atomic_add_f32_native(&deg[dst[e]], 1.0f);
}

__global__ void rsqrt_kernel(float* __restrict__ p, int n) {
  int i = blockIdx.x * blockDim.x + threadIdx.x;
  if (i < n) p[i] = rsqrtf(p[i]);
}

// ------------------------------------------------------------------ BN stats
// 128 threads/block: thread c accumulates channel c over a row stripe, then
// f32-atomics into stats[0:128]=sum, stats[128:256]=sumsq.
__global__ void stats_kernel(const float* __restrict__ x, float* __restrict__ stats,
                             int nRows, int doRelu) {
  int c = threadIdx.x;  // 0..127
  float s = 0.0f, s2 = 0.0f;
  for (int r = blockIdx.x; r < nRows; r += gridDim.x) {
    float v = x[(size_t)r * CH + c];
    if (doRelu) v = fmaxf(v, 0.0f);
    s += v;
    s2 += v * v;
  }
  atomic_add_f32_native(&stats[c], s);
  atomic_add_f32_native(&stats[CH + c], s2);
}

// mean/var -> per-channel scale s and shift t:  BN(x)_c = x_c*s_c + t_c
__global__ void bn_finalize_kernel(const float* __restrict__ stats,
                                   const float* __restrict__ gamma,
                                   const float* __restrict__ beta,
                                   float* __restrict__ s_out, float* __restrict__ t_out,
                                   float invN) {
  int c = threadIdx.x;  // 128 threads, 1 block
  float mean = stats[c] * invN;
  float var = stats[CH + c] * invN - mean * mean;
  float sc = gamma[c] * rsqrtf(var + BN_EPS);
  s_out[c] = sc;
  t_out[c] = beta[c] - mean * sc;
}

// Fold BN into GEMM:  Wf[k,c] = W[k,c]*s[c],  tb[k] = sum_c t[c]*W[k,c]
__global__ void fold_w_kernel(const float* __restrict__ W, const float* __restrict__ s,
                              const float* __restrict__ t, float* __restrict__ Wf,
                              float* __restrict__ tb) {
  int k = threadIdx.x;  // 128 threads, 1 block
  float acc = 0.0f;
  for (int c = 0; c < CH; ++c) {
    float w = W[k * CH + c];
    Wf[k * CH + c] = w * s[c];
    acc += w * t[c];
  }
  tb[k] = acc;
}

// --------------------------------------------------------------- WMMA GEMM
// out[N,128] = (relu?)(A[N,128]) @ Wf[128,128]^T + tb
// Block = 256 threads = 8 waves. Block covers 16 rows; wave w covers cols
// [16w, 16w+16). K accumulated with 32x v_wmma_f32_16x16x4_f32 (exact f32).
__global__ void gemm_bn_kernel(const float* __restrict__ A, const float* __restrict__ Wf,
                               const float* __restrict__ tb, float* __restrict__ out,
                               int nRows, int doRelu) {
  __shared__ __align__(16) float ldsA[16 * CH];
  int rowBase = blockIdx.x * 16;
  int tid = threadIdx.x;

  // Stage the 16x128 A tile through LDS (optionally fused ReLU).
  for (int i = tid; i < 16 * CH; i += 256) {
    int r = i >> 7, c = i & (CH - 1);
    int row = rowBase + r;
    float v = (row < nRows) ? A[(size_t)row * CH + c] : 0.0f;
    if (doRelu) v = fmaxf(v, 0.0f);
    ldsA[i] = v;
  }
  __syncthreads();

  int wave = tid >> 5;      // 0..7 -> 16-col slab
  int lane = tid & 31;
  int lh = lane & 15;       // M (A) or N (B) index within tile
  int hi = lane >> 4;       // selects K pair {0,1} vs {2,3}
  int colBase = wave * 16;

  const float* arow = ldsA + lh * CH + 2 * hi;                    // A: row lh
  const float* brow = Wf + (size_t)(colBase + lh) * CH + 2 * hi;  // B[k,n] = Wf[n,k]

  v8f acc = {};
  for (int kk = 0; kk < CH; kk += 4) {
    v2f a = *(const v2f*)(arow + kk);
    v2f b = *(const v2f*)(brow + kk);
    acc = __builtin_amdgcn_wmma_f32_16x16x4_f32(
        /*neg_a=*/false, a, /*neg_b=*/false, b,
        /*c_mod=*/(short)0, acc, /*reuse_a=*/false, /*reuse_b=*/false);
  }

  float tbv = tb[colBase + lh];
#pragma unroll
  for (int r = 0; r < 8; ++r) {
    int row = rowBase + r + 8 * hi;  // C/D layout: VGPR r -> M=r (lo half), r+8 (hi half)
    if (row < nRows) out[(size_t)row * CH + colBase + lh] = acc[r] + tbv;
  }
}

// -------------------------------------------------------- propagate (A+I) D^-1/2
// Self-loop + bias init: out[n,c] = dis[n]^2 * h[n,c] + b[c]
__global__ void self_loop_init_kernel(const float* __restrict__ h, const float* __restrict__ dis,
                                      const float* __restrict__ b, float* __restrict__ out,
                                      int nElem) {
  int i = blockIdx.x * blockDim.x + threadIdx.x;
  if (i >= nElem) return;
  int n = i >> 7, c = i & (CH - 1);
  float d = dis[n];
  out[i] = d * d * h[i] + b[c];
}

// One wave per edge; lane handles 4 channels: float4 gather + 4 native atomics.
__global__ void edge_scatter_kernel(const float* __restrict__ h, const int* __restrict__ src,
                                    const int* __restrict__ dst, const float* __restrict__ dis,
                                    float* __restrict__ out, int nEdges) {
  int e = (blockIdx.x * blockDim.x + threadIdx.x) >> 5;
  if (e >= nEdges) return;
  int lane = threadIdx.x & 31;
  int s = src[e], d = dst[e];
  float norm = dis[s] * dis[d];
  float4 v = ((const float4*)(h + (size_t)s * CH))[lane];
  float* op = out + (size_t)d * CH + lane * 4;
  atomic_add_f32_native(op + 0, norm * v.x);
  atomic_add_f32_native(op + 1, norm * v.y);
  atomic_add_f32_native(op + 2, norm * v.z);
  atomic_add_f32_native(op + 3, norm * v.w);
}

// out = relu(acc) + x
__global__ void final_kernel(const float* __restrict__ acc, const float* __restrict__ x,
                             float* __restrict__ out, int nElem) {
  int i = blockIdx.x * blockDim.x + threadIdx.x;
  if (i < nElem) out[i] = fmaxf(acc[i], 0.0f) + x[i];
}

// ------------------------------------------------------------------- driver
extern "C" void kernel_launch(void* const* d_in, const int* in_sizes, int n_in,
                              void* d_out, int out_size, void* d_ws, size_t ws_size,
                              hipStream_t stream) {
  const float* x      = (const float*)d_in[0];
  const int*   ei     = (const int*)d_in[1];
  const float* W1     = (const float*)d_in[2];
  const float* b1     = (const float*)d_in[3];
  const float* W2     = (const float*)d_in[4];
  const float* b2     = (const float*)d_in[5];
  const float* gamma1 = (const float*)d_in[6];
  const float* beta1  = (const float*)d_in[7];
  const float* gamma2 = (const float*)d_in[8];
  const float* beta2  = (const float*)d_in[9];

  const int N = in_sizes[0] / CH;   // 100000
  const int E = in_sizes[1] / 2;    // 1000000
  const int NC = N * CH;
  const int* src = ei;
  const int* dst = ei + E;

  // Workspace layout (floats), ~103 MB total.
  float* bufA  = (float*)d_ws;                 // h1 / h2        (NC)
  float* bufB  = bufA + NC;                    // out1 / out2    (NC)
  float* deg   = bufB + NC;                    // deg -> deg^-1/2 (N)
  float* Wf    = deg + N;                      // folded weights (CH*CH)
  float* tb    = Wf + CH * CH;                 // folded offsets (CH)
  float* stats = tb + CH;                      // sum|sumsq      (2*CH)
  float* sbn   = stats + 2 * CH;               // BN scale       (CH)
  float* tbn   = sbn + CH;                     // BN shift       (CH)

  const float invN = 1.0f / (float)N;
  const int TB = 256;
  dim3 blk(TB);
  dim3 gN((N + TB - 1) / TB);
  dim3 gE((E + TB - 1) / TB);
  dim3 gNC((NC + TB - 1) / TB);
  dim3 gGemm(N / 16 + (N % 16 ? 1 : 0));
  dim3 gScat((E * 32 + TB - 1) / TB);

  // ---- normalization coefficients: deg (with self loop) -> rsqrt
  fill_kernel<<<gN, blk, 0, stream>>>(deg, 1.0f, N);
  degree_kernel<<<gE, blk, 0, stream>>>(dst, deg, E);
  rsqrt_kernel<<<gN, blk, 0, stream>>>(deg, N);

  // ---- layer 1: BN1 folded into W1, WMMA GEMM, propagate
  fill_kernel<<<1, blk, 0, stream>>>(stats, 0.0f, 2 * CH);
  stats_kernel<<<512, dim3(CH), 0, stream>>>(x, stats, N, /*relu=*/0);
  bn_finalize_kernel<<<1, dim3(CH), 0, stream>>>(stats, gamma1, beta1, sbn, tbn, invN);
  fold_w_kernel<<<1, dim3(CH), 0, stream>>>(W1, sbn, tbn, Wf, tb);
  gemm_bn_kernel<<<gGemm, blk, 0, stream>>>(x, Wf, tb, bufA, N, /*relu=*/0);
  self_loop_init_kernel<<<gNC, blk, 0, stream>>>(bufA, deg, b1, bufB, NC);
  edge_scatter_kernel<<<gScat, blk, 0, stream>>>(bufA, src, dst, deg, bufB, E);

  // ---- layer 2: BN2(relu(out1)) folded into W2, WMMA GEMM (relu fused), propagate
  fill_kernel<<<1, blk, 0, stream>>>(stats, 0.0f, 2 * CH);
  stats_kernel<<<512, dim3(CH), 0, stream>>>(bufB, stats, N, /*relu=*/1);
  bn_finalize_kernel<<<1, dim3(CH), 0, stream>>>(stats, gamma2, beta2, sbn, tbn, invN);
  fold_w_kernel<<<1, dim3(CH), 0, stream>>>(W2, sbn, tbn, Wf, tb);
  gemm_bn_kernel<<<gGemm, blk, 0, stream>>>(bufB, Wf, tb, bufA, N, /*relu=*/1);
  self_loop_init_kernel<<<gNC, blk, 0, stream>>>(bufA, deg, b2, bufB, NC);
  edge_scatter_kernel<<<gScat, blk, 0, stream>>>(bufA, src, dst, deg, bufB, E);

  // ---- residual + relu
  final_kernel<<<gNC, blk, 0, stream>>>(bufB, x, (float*)d_out, NC);
}